// MagicModel_83562883711405
// MI455X (gfx1250) — compile-verified
//
#include <hip/hip_runtime.h>
#include <hip/hip_bf16.h>

#define N_NODES 100000
#define N_EDGES 3200000
#define N_GRAPHS 256
#define FDIM 200
#define CONCAT 1000
#define K1PAD 12     // layer-1 input dim 11 padded to multiple of 4
#define NPAD 208     // layer output dim 200 padded to multiple of 16

typedef __attribute__((ext_vector_type(2))) float v2f;
typedef __attribute__((ext_vector_type(8))) float v8f;

// ---------------------------------------------------------------------------
// 2-D pad-copy: dst[r, c] = (r < rin && c < cin) ? src[r, c] : 0
__global__ void pad2_kernel(const float* __restrict__ src, float* __restrict__ dst,
                            int rin, int cin, int rpad, int cpad) {
    long long idx = (long long)blockIdx.x * blockDim.x + threadIdx.x;
    long long total = (long long)rpad * cpad;
    if (idx >= total) return;
    int r = (int)(idx / cpad);
    int c = (int)(idx - (long long)r * cpad);
    dst[idx] = (r < rin && c < cin) ? src[(long long)r * cin + c] : 0.f;
}

// degree: deg[dst] += 1 per edge
__global__ void deg_kernel(const int* __restrict__ dst, float* __restrict__ deg, int E) {
    long long i = (long long)blockIdx.x * blockDim.x + threadIdx.x;
    if (i < E) atomicAdd(&deg[dst[i]], 1.0f);
}

__global__ void invdeg_kernel(const float* __restrict__ deg, float* __restrict__ invdeg, int n) {
    long long i = (long long)blockIdx.x * blockDim.x + threadIdx.x;
    if (i < n) invdeg[i] = 1.0f / fmaxf(deg[i], 1.0f);
}

// ---------------------------------------------------------------------------
// agg[dst, f..f+3] += h[src*sH + coH + f..f+3]   (F divisible by 4)
__global__ void scatter4_kernel(const float* __restrict__ h,
                                const int* __restrict__ src,
                                const int* __restrict__ dst,
                                float* __restrict__ agg,
                                int E, int F, int sH, int coH) {
    const int Fq = F >> 2;
    long long idx = (long long)blockIdx.x * blockDim.x + threadIdx.x;
    long long total = (long long)E * Fq;
    if (idx >= total) return;
    int e  = (int)(idx / Fq);
    int f  = (int)(idx - (long long)e * Fq) << 2;
    int s  = src[e];
    int d  = dst[e];
    const float4 v = *(const float4*)(h + (long long)s * sH + coH + f);
    float* ap = agg + (long long)d * F + f;
    atomicAdd(ap + 0, v.x);
    atomicAdd(ap + 1, v.y);
    atomicAdd(ap + 2, v.z);
    atomicAdd(ap + 3, v.w);
}

// agg[n, f] *= invdeg[n]
__global__ void scale_kernel(float* __restrict__ agg, const float* __restrict__ invdeg,
                             long long total, int F) {
    long long idx = (long long)blockIdx.x * blockDim.x + threadIdx.x;
    if (idx >= total) return;
    agg[idx] *= invdeg[idx / F];
}

// ---------------------------------------------------------------------------
// fp32 WMMA GEMM, MT row-tiles per wave, software-pipelined (double-buffered):
//   out[m, coO+n] = act( A1[m,:]@Wl[n,:] + bias[n] (+ A2[m,:]@Wr[n,:]) )
// Wl/Wr row-padded to gridDim.y*16 rows (zero rows beyond Nout).
// M % (16*MT) == 0, K % 4 == 0, all strides even (8B-aligned v2f loads).
template <int MT, bool DUAL, bool RELU>
__global__ void __launch_bounds__(32)
wmma_gemm(const float* __restrict__ A1, int sA1,
          const float* __restrict__ A2, int sA2, int coA2,
          const float* __restrict__ Wlp, const float* __restrict__ Wrp,
          const float* __restrict__ bias,
          int K, int Nout,
          float* __restrict__ out, int sO, int coO) {
    const int lane = threadIdx.x;          // wave32
    const int row0 = blockIdx.x * (16 * MT);
    const int n0   = blockIdx.y << 4;
    const int ml   = lane & 15;
    const int kb   = (lane >> 4) << 1;     // K-pair select per ISA A/B layout

    const int ncol = n0 + ml;              // always a valid (possibly zero) W row

    v8f c[MT];
#pragma unroll
    for (int t = 0; t < MT; ++t) c[t] = (v8f){};

    const float* ap1[MT];
    const float* ap2[MT];
#pragma unroll
    for (int t = 0; t < MT; ++t) {
        ap1[t] = A1 + (long long)(row0 + 16 * t + ml) * sA1 + kb;
        if (DUAL) ap2[t] = A2 + (long long)(row0 + 16 * t + ml) * sA2 + coA2 + kb;
    }
    const float* wlp = Wlp + (long long)ncol * K + kb;
    const float* wrp = DUAL ? (Wrp + (long long)ncol * K + kb) : nullptr;

    // ---- prologue: load k=0 fragments (B first, then A tiles) ----
    v2f blc = *(const v2f*)(wlp);
    v2f brc;
    if (DUAL) brc = *(const v2f*)(wrp);
    v2f a1c[MT], a2c[MT];
#pragma unroll
    for (int t = 0; t < MT; ++t) a1c[t] = *(const v2f*)(ap1[t]);
    if (DUAL) {
#pragma unroll
        for (int t = 0; t < MT; ++t) a2c[t] = *(const v2f*)(ap2[t]);
    }

    // ---- steady state: issue next k-step loads, then current WMMAs ----
    // unroll 1: double-buffer registers cover exactly one iteration ahead;
    // a x2 unroll serializes the second copy (not enough VGPRs to pipeline two).
#pragma unroll 1
    for (int k0 = 0; k0 + 4 < K; k0 += 4) {
        const int kn = k0 + 4;
        __builtin_prefetch(ap1[0] + kn + 60, 0, 3);

        v2f bln = *(const v2f*)(wlp + kn);
        v2f brn;
        if (DUAL) brn = *(const v2f*)(wrp + kn);
        v2f a1n[MT], a2n[MT];
#pragma unroll
        for (int t = 0; t < MT; ++t) a1n[t] = *(const v2f*)(ap1[t] + kn);
        if (DUAL) {
#pragma unroll
            for (int t = 0; t < MT; ++t) a2n[t] = *(const v2f*)(ap2[t] + kn);
        }

#pragma unroll
        for (int t = 0; t < MT; ++t)
            c[t] = __builtin_amdgcn_wmma_f32_16x16x4_f32(false, a1c[t], false, blc,
                                                         (short)0, c[t], false, false);
        if (DUAL) {
#pragma unroll
            for (int t = 0; t < MT; ++t)
                c[t] = __builtin_amdgcn_wmma_f32_16x16x4_f32(false, a2c[t], false, brc,
                                                             (short)0, c[t], false, false);
        }

        blc = bln;
        if (DUAL) brc = brn;
#pragma unroll
        for (int t = 0; t < MT; ++t) {
            a1c[t] = a1n[t];
            if (DUAL) a2c[t] = a2n[t];
        }
    }

    // ---- epilogue: drain last k-step ----
#pragma unroll
    for (int t = 0; t < MT; ++t)
        c[t] = __builtin_amdgcn_wmma_f32_16x16x4_f32(false, a1c[t], false, blc,
                                                     (short)0, c[t], false, false);
    if (DUAL) {
#pragma unroll
        for (int t = 0; t < MT; ++t)
            c[t] = __builtin_amdgcn_wmma_f32_16x16x4_f32(false, a2c[t], false, brc,
                                                         (short)0, c[t], false, false);
    }

    if (ncol < Nout) {
        const float bv = bias ? bias[ncol] : 0.f;
#pragma unroll
        for (int t = 0; t < MT; ++t) {
            const int mbase = row0 + 16 * t + ((lane >> 4) << 3);  // lanes 16-31: M+8
#pragma unroll
            for (int r = 0; r < 8; ++r) {
                float v = c[t][r] + bv;
                if (RELU) v = fmaxf(v, 0.f);
                out[(long long)(mbase + r) * sO + coO + ncol] = v;
            }
        }
    }
}

// ---------------------------------------------------------------------------
__global__ void cnt_kernel(const int* __restrict__ batch, float* __restrict__ cnt, int n) {
    long long i = (long long)blockIdx.x * blockDim.x + threadIdx.x;
    if (i < n) atomicAdd(&cnt[batch[i]], 1.0f);
}

// gsum[batch[n], f..f+3] += hc[n, f..f+3]
__global__ void pool_scatter4_kernel(const float* __restrict__ hc, const int* __restrict__ batch,
                                     float* __restrict__ gsum, int n, int F) {
    const int Fq = F >> 2;
    long long idx = (long long)blockIdx.x * blockDim.x + threadIdx.x;
    long long total = (long long)n * Fq;
    if (idx >= total) return;
    int node = (int)(idx / Fq);
    int f    = (int)(idx - (long long)node * Fq) << 2;
    const float4 v = *(const float4*)(hc + (long long)node * F + f);
    float* gp = gsum + (long long)batch[node] * F + f;
    atomicAdd(gp + 0, v.x);
    atomicAdd(gp + 1, v.y);
    atomicAdd(gp + 2, v.z);
    atomicAdd(gp + 3, v.w);
}

__global__ void pool_div_kernel(float* __restrict__ gsum, const float* __restrict__ cnt,
                                int G, int F) {
    long long idx = (long long)blockIdx.x * blockDim.x + threadIdx.x;
    if (idx >= (long long)G * F) return;
    gsum[idx] /= fmaxf(cnt[idx / F], 1.0f);
}

// final 250 -> 1 + sigmoid
__global__ void head_final_kernel(const float* __restrict__ g2, const float* __restrict__ pw3,
                                  const float* __restrict__ pb3, float* __restrict__ out,
                                  int G, int K) {
    int b = blockIdx.x * blockDim.x + threadIdx.x;
    if (b >= G) return;
    float acc = pb3[0];
    for (int k = 0; k < K; ++k) acc += g2[(long long)b * K + k] * pw3[k];
    out[b] = 1.0f / (1.0f + expf(-acc));
}

// ---------------------------------------------------------------------------
extern "C" void kernel_launch(void* const* d_in, const int* in_sizes, int n_in,
                              void* d_out, int out_size, void* d_ws, size_t ws_size,
                              hipStream_t stream) {
    (void)in_sizes; (void)n_in; (void)out_size; (void)ws_size;

    const float* x     = (const float*)d_in[0];
    const int*   eidx  = (const int*)d_in[1];
    const int*   batch = (const int*)d_in[2];
    const int*   esrc  = eidx;
    const int*   edst  = eidx + N_EDGES;

    const float* Wl[5], * bl[5], * Wr[5];
    for (int i = 0; i < 5; ++i) {
        Wl[i] = (const float*)d_in[3 + 3 * i];
        bl[i] = (const float*)d_in[4 + 3 * i];
        Wr[i] = (const float*)d_in[5 + 3 * i];
    }
    const float* pw1 = (const float*)d_in[18];
    const float* pb1 = (const float*)d_in[19];
    const float* pw2 = (const float*)d_in[20];
    const float* pb2 = (const float*)d_in[21];
    const float* pw3 = (const float*)d_in[22];
    const float* pb3 = (const float*)d_in[23];
    float* out = (float*)d_out;

    // workspace carve-out (floats, 256B aligned)
    float* ws = (float*)d_ws;
    size_t off = 0;
    auto take = [&](size_t n) { float* p = ws + off; off += (n + 63) & ~(size_t)63; return p; };
    float* deg    = take((size_t)N_NODES);
    float* invdeg = take((size_t)N_NODES);
    float* cnt    = take((size_t)N_GRAPHS);
    float* xpad   = take((size_t)N_NODES * K1PAD);         // x: 11 -> 12 cols
    float* WlP[5], * WrP[5];
    WlP[0] = take((size_t)NPAD * K1PAD);                   // 208 x 12
    WrP[0] = take((size_t)NPAD * K1PAD);
    for (int i = 1; i < 5; ++i) {
        WlP[i] = take((size_t)NPAD * FDIM);                // 208 x 200
        WrP[i] = take((size_t)NPAD * FDIM);
    }
    float* pw1p   = take((size_t)512 * CONCAT);            // 512 x 1000
    float* pw2p   = take((size_t)256 * 500);               // 256 x 500
    float* agg    = take((size_t)N_NODES * FDIM);
    float* hc     = take((size_t)N_NODES * CONCAT);
    float* gsum   = take((size_t)N_GRAPHS * CONCAT);
    float* g1     = take((size_t)N_GRAPHS * 500);
    float* g2     = take((size_t)N_GRAPHS * 250);

    const int T = 256;
    auto blocks = [](long long n, int t) { return (unsigned)((n + t - 1) / t); };

    // stage padded operands (zero pad rows/cols)
    pad2_kernel<<<blocks((long long)N_NODES * K1PAD, T), T, 0, stream>>>(
        x, xpad, N_NODES, 11, N_NODES, K1PAD);
    pad2_kernel<<<blocks((long long)NPAD * K1PAD, T), T, 0, stream>>>(
        Wl[0], WlP[0], FDIM, 11, NPAD, K1PAD);
    pad2_kernel<<<blocks((long long)NPAD * K1PAD, T), T, 0, stream>>>(
        Wr[0], WrP[0], FDIM, 11, NPAD, K1PAD);
    for (int i = 1; i < 5; ++i) {
        pad2_kernel<<<blocks((long long)NPAD * FDIM, T), T, 0, stream>>>(
            Wl[i], WlP[i], FDIM, FDIM, NPAD, FDIM);
        pad2_kernel<<<blocks((long long)NPAD * FDIM, T), T, 0, stream>>>(
            Wr[i], WrP[i], FDIM, FDIM, NPAD, FDIM);
    }
    pad2_kernel<<<blocks((long long)512 * CONCAT, T), T, 0, stream>>>(
        pw1, pw1p, 500, CONCAT, 512, CONCAT);
    pad2_kernel<<<blocks((long long)256 * 500, T), T, 0, stream>>>(
        pw2, pw2p, 250, 500, 256, 500);

    // degree (shared across layers)
    hipMemsetAsync(deg, 0, (size_t)N_NODES * sizeof(float), stream);
    deg_kernel<<<blocks(N_EDGES, T), T, 0, stream>>>(edst, deg, N_EDGES);
    invdeg_kernel<<<blocks(N_NODES, T), T, 0, stream>>>(deg, invdeg, N_NODES);

    // 5 SAGE layers, outputs written into column slices of hc [N, 1000]
    for (int l = 0; l < 5; ++l) {
        const int Fin    = (l == 0) ? K1PAD : FDIM;   // padded K, % 4 == 0
        const float* hin = (l == 0) ? xpad : hc;
        const int sH     = (l == 0) ? K1PAD : CONCAT;
        const int coH    = (l == 0) ? 0 : (l - 1) * FDIM;

        hipMemsetAsync(agg, 0, (size_t)N_NODES * Fin * sizeof(float), stream);
        scatter4_kernel<<<blocks((long long)N_EDGES * (Fin >> 2), T), T, 0, stream>>>(
            hin, esrc, edst, agg, N_EDGES, Fin, sH, coH);
        scale_kernel<<<blocks((long long)N_NODES * Fin, T), T, 0, stream>>>(
            agg, invdeg, (long long)N_NODES * Fin, Fin);

        // 100000 = 1250 * (16*5); 13 column tiles cover NPAD=208 weight rows
        dim3 grid(N_NODES / (16 * 5), NPAD / 16);
        wmma_gemm<5, true, true><<<grid, 32, 0, stream>>>(
            agg, Fin, hin, sH, coH,
            WlP[l], WrP[l], bl[l],
            Fin, FDIM,
            hc, CONCAT, l * FDIM);
    }

    // global mean pool
    hipMemsetAsync(cnt, 0, (size_t)N_GRAPHS * sizeof(float), stream);
    hipMemsetAsync(gsum, 0, (size_t)N_GRAPHS * CONCAT * sizeof(float), stream);
    cnt_kernel<<<blocks(N_NODES, T), T, 0, stream>>>(batch, cnt, N_NODES);
    pool_scatter4_kernel<<<blocks((long long)N_NODES * (CONCAT >> 2), T), T, 0, stream>>>(
        hc, batch, gsum, N_NODES, CONCAT);
    pool_div_kernel<<<blocks((long long)N_GRAPHS * CONCAT, T), T, 0, stream>>>(
        gsum, cnt, N_GRAPHS, CONCAT);

    // projection head (reference has no activations between head layers)
    {
        dim3 grid1(N_GRAPHS / 16, 512 / 16);
        wmma_gemm<1, false, false><<<grid1, 32, 0, stream>>>(
            gsum, CONCAT, nullptr, 0, 0,
            pw1p, nullptr, pb1,
            CONCAT, 500, g1, 500, 0);
        dim3 grid2(N_GRAPHS / 16, 256 / 16);
        wmma_gemm<1, false, false><<<grid2, 32, 0, stream>>>(
            g1, 500, nullptr, 0, 0,
            pw2p, nullptr, pb2,
            500, 250, g2, 250, 0);
    }
    head_final_kernel<<<1, 256, 0, stream>>>(g2, pw3, pb3, out, N_GRAPHS, 250);
}